// GraphSAGE_23227183137262
// MI455X (gfx1250) — compile-verified
//
#include <hip/hip_runtime.h>
#include <stdint.h>

// GraphSAGE 2-layer forward for MI455X (gfx1250, wave32).
// Phase 1/3: edge scatter (memory/atomic bound, ~1.7 GB total traffic).
// Phase 2/4: dense SAGE transform using V_WMMA_F32_16X16X4_F32 (full fp32
//            precision; GEMM work is only ~3.3 GFLOP so fp32 WMMA is free).

typedef __attribute__((ext_vector_type(2))) float v2f;
typedef __attribute__((ext_vector_type(8))) float v8f;

#define WAVE 32

// ---------------------------------------------------------------------------
// Edge scatter: agg[dst] += feat[src]  (+ deg[dst] += 1 when deg != nullptr)
// 4 threads per edge, 16 floats (4x float4) each.
// ---------------------------------------------------------------------------
__global__ __launch_bounds__(256) void sage_scatter(
    const float* __restrict__ feat, const long long* __restrict__ src,
    const long long* __restrict__ dst, float* __restrict__ agg,
    float* __restrict__ deg, int nedges)
{
  int idx = blockIdx.x * blockDim.x + threadIdx.x;
  int e = idx >> 2;
  if (e >= nedges) return;
  int q = idx & 3;
  long long s = src[e];
  long long d = dst[e];
  const float4* xs = (const float4*)(feat + s * 64 + q * 16);
  float* ad = agg + d * 64 + q * 16;
#pragma unroll
  for (int i = 0; i < 4; ++i) {
    float4 v = xs[i];
    atomicAdd(ad + 4 * i + 0, v.x);
    atomicAdd(ad + 4 * i + 1, v.y);
    atomicAdd(ad + 4 * i + 2, v.z);
    atomicAdd(ad + 4 * i + 3, v.w);
  }
  if (deg != nullptr && q == 0) atomicAdd(deg + d, 1.0f);
}

// ---------------------------------------------------------------------------
// Dense SAGE layer:  out = act( (agg/deg) @ Wl^T + b + xin @ Wr^T )
// One wave per 16-node tile; NTILES 16-wide output column tiles.
// A fragment (16x4 f32): lane L -> row L%16, K pair 2*(L/16).
// B fragment (4x16 f32): lane L -> col L%16, K pair 2*(L/16) (B[K][N]=W[N][K]).
// C/D (16x16 f32): lane L, VGPR v -> [M = v + 8*(L/16)][N = L%16].
// ---------------------------------------------------------------------------
template <int NTILES, bool RELU>
__global__ __launch_bounds__(256) void sage_dense(
    const float* __restrict__ aggsum, const float* __restrict__ deg,
    const float* __restrict__ xin, const float* __restrict__ Wl,
    const float* __restrict__ Wr, const float* __restrict__ bias,
    float* __restrict__ out, int nnodes, int outf, int ostride)
{
  __shared__ float sWl[NTILES * 16 * 64];
  __shared__ float sWr[NTILES * 16 * 64];
  __shared__ float sB[NTILES * 16];

  const int tid = threadIdx.x;
  // Stage weights into LDS, zero-padding rows >= outf.
  for (int i = tid; i < NTILES * 16 * 64; i += blockDim.x) {
    int r = i >> 6;
    bool ok = (r < outf);
    sWl[i] = ok ? Wl[i] : 0.0f;
    sWr[i] = ok ? Wr[i] : 0.0f;
  }
  for (int i = tid; i < NTILES * 16; i += blockDim.x)
    sB[i] = (i < outf) ? bias[i] : 0.0f;
  __syncthreads();

  const int wave = tid / WAVE;
  const int lane = tid % WAVE;
  const int tile = blockIdx.x * (blockDim.x / WAVE) + wave;
  const int node_tiles = (nnodes + 15) >> 4;
  if (tile >= node_tiles) return;  // wave-uniform exit, EXEC stays full

  const int m = lane & 15;       // A row / B-D column within tile
  const int halfk = lane >> 4;   // which K pair this lane supplies
  const int koff = halfk * 2;

  long long row = (long long)tile * 16 + m;
  if (row >= nnodes) row = nnodes - 1;  // clamp loads, keep EXEC all-1s
  const float dv = deg[row];
  const float inv = 1.0f / fmaxf(dv, 1.0f);
  const float* arow = aggsum + row * 64;
  const float* xrow = xin + row * 64;

  v8f acc[NTILES];
#pragma unroll
  for (int t = 0; t < NTILES; ++t) {
    float bv = sB[t * 16 + m];  // every VGPR of this lane has N = m
    acc[t] = (v8f){bv, bv, bv, bv, bv, bv, bv, bv};
  }

#pragma unroll
  for (int k = 0; k < 64; k += 4) {
    v2f a_agg = *(const v2f*)(arow + k + koff);
    a_agg *= inv;  // mean aggregation
    v2f a_x = *(const v2f*)(xrow + k + koff);
#pragma unroll
    for (int t = 0; t < NTILES; ++t) {
      const int ncol = t * 16 + m;
      v2f bl = *(const v2f*)(&sWl[ncol * 64 + k + koff]);
      v2f br = *(const v2f*)(&sWr[ncol * 64 + k + koff]);
      acc[t] = __builtin_amdgcn_wmma_f32_16x16x4_f32(
          false, a_agg, false, bl, (short)0, acc[t], false, false);
      acc[t] = __builtin_amdgcn_wmma_f32_16x16x4_f32(
          false, a_x, false, br, (short)0, acc[t], false, false);
    }
  }

  // Store: lane L, VGPR v covers node row (tile*16 + 8*halfk + v), col t*16+m.
  const long long obase = (long long)tile * 16 + halfk * 8;
#pragma unroll
  for (int t = 0; t < NTILES; ++t) {
    const int ncol = t * 16 + m;
    if (ncol < outf) {
#pragma unroll
      for (int v = 0; v < 8; ++v) {
        long long node = obase + v;
        if (node < nnodes) {
          float val = acc[t][v];
          if (RELU) val = fmaxf(val, 0.0f);
          out[node * (long long)ostride + ncol] = val;
        }
      }
    }
  }
}

// ---------------------------------------------------------------------------
// Launch: inputs (dict order) = x, edge_index, W1_l, b1, W1_r, W2_l, b2, W2_r
// ---------------------------------------------------------------------------
extern "C" void kernel_launch(void* const* d_in, const int* in_sizes, int n_in,
                              void* d_out, int out_size, void* d_ws,
                              size_t ws_size, hipStream_t stream)
{
  const float* x = (const float*)d_in[0];
  const long long* ei = (const long long*)d_in[1];
  const float* W1l = (const float*)d_in[2];
  const float* b1 = (const float*)d_in[3];
  const float* W1r = (const float*)d_in[4];
  const float* W2l = (const float*)d_in[5];
  const float* b2 = (const float*)d_in[6];
  const float* W2r = (const float*)d_in[7];
  float* out = (float*)d_out;

  const int N = in_sizes[0] / 64;  // 100000
  const int E = in_sizes[1] / 2;   // 1600000
  const long long* src = ei;       // edge_index[0]
  const long long* dst = ei + E;   // edge_index[1]

  // Workspace: agg1[N*64] | deg[N] | h[N*64] | agg2[N*64]  (~77.2 MB)
  float* agg1 = (float*)d_ws;
  float* degw = agg1 + (size_t)N * 64;
  float* h = degw + N;
  float* agg2 = h + (size_t)N * 64;

  // Deterministic zero-init every call (memset nodes are graph-capturable).
  hipMemsetAsync(agg1, 0, (size_t)(N * 64 + N) * sizeof(float), stream);
  hipMemsetAsync(agg2, 0, (size_t)N * 64 * sizeof(float), stream);

  const int sblocks = (E * 4 + 255) / 256;
  const int tiles = (N + 15) / 16;
  const int dblocks = (tiles + 7) / 8;  // 8 waves (tiles) per 256-thread block

  // Layer 1: aggregate x, then h = relu(mean @ W1l^T + b1 + x @ W1r^T)
  sage_scatter<<<sblocks, 256, 0, stream>>>(x, src, dst, agg1, degw, E);
  sage_dense<4, true><<<dblocks, 256, 0, stream>>>(agg1, degw, x, W1l, W1r, b1,
                                                   h, N, 64, 64);

  // Layer 2: aggregate h (degree unchanged), out = mean @ W2l^T + b2 + h @ W2r^T
  sage_scatter<<<sblocks, 256, 0, stream>>>(h, src, dst, agg2, nullptr, E);
  sage_dense<3, false><<<dblocks, 256, 0, stream>>>(agg2, degw, h, W2l, W2r,
                                                    b2, out, N, 40, 40);
}